// VisualRelationExplainer_63196148793818
// MI455X (gfx1250) — compile-verified
//
#include <hip/hip_runtime.h>
#include <hip/hip_bf16.h>
#include <math.h>

// ---------------- problem constants ----------------
#define BGRAPH   64
#define NPG      50
#define EPG      2048
#define NNODE    (BGRAPH * NPG)      // 3200
#define NEDGE    (BGRAPH * EPG)      // 131072
#define D_LIN    1024
#define D_CONV   12544               // 256*7*7
#define D_REL    256
#define PCLS     50
#define PPAD     64                  // P padded to 4 WMMA N-tiles
#define KTOP     10
#define XTOP     100

#define NCHUNK   64                  // K-chunk staged in LDS (node kernel)
#define NSTRIDE  (NCHUNK + 4)        // padded LDS row stride -> conflict-free
#define ESTRIDE  (D_REL + 4)         // edge kernel stages whole K=256 panel

typedef float v2f __attribute__((ext_vector_type(2)));
typedef float v8f __attribute__((ext_vector_type(8)));

// ---- order-preserving float<->uint encoding (for atomic max with argmax) ----
__device__ __forceinline__ unsigned f2ord(float f) {
    unsigned u = __float_as_uint(f);
    return (u & 0x80000000u) ? ~u : (u | 0x80000000u);
}
__device__ __forceinline__ float ord2f(unsigned o) {
    unsigned u = (o & 0x80000000u) ? (o & 0x7FFFFFFFu) : ~o;
    return __uint_as_float(u);
}

// ---- gfx1250 async copy: global -> LDS, 16B per lane, tracked by ASYNCcnt ----
__device__ __forceinline__ void async_g2l_b128(unsigned lds_off, const float* gptr) {
    asm volatile("global_load_async_to_lds_b128 %0, %1, off"
                 :: "v"(lds_off), "v"((unsigned long long)(uintptr_t)gptr)
                 : "memory");
}
__device__ __forceinline__ void wait_asynccnt0() {
    asm volatile("s_wait_asynccnt 0x0" ::: "memory");
}

// ---------------- kernel: zero the (graph, class) argmax slots ----------------
__global__ void init_enc_kernel(unsigned long long* __restrict__ enc, int n) {
    int i = blockIdx.x * blockDim.x + threadIdx.x;
    if (i < n) enc[i] = 0ull;   // 0 encodes "below -inf"
}

// ---------------- kernel: pad weights D x 50 -> D x 64 (zero fill) ------------
__global__ void pad_weights_kernel(const float* __restrict__ W,
                                   float* __restrict__ Wp, int total) {
    int i = blockIdx.x * blockDim.x + threadIdx.x;
    if (i >= total) return;
    int d = i >> 6, n = i & 63;
    Wp[i] = (n < PCLS) ? W[d * PCLS + n] : 0.0f;
}

// ---------------- kernel: per-class L1 norms of weight columns ---------------
__global__ void l1_kernel(const float* __restrict__ Wo, const float* __restrict__ Wc,
                          const float* __restrict__ Wr,
                          float* __restrict__ L1n, float* __restrict__ L1r) {
    const int c = blockIdx.x;           // 0..49
    const int t = threadIdx.x;          // 256 threads
    float sn = 0.0f, sr = 0.0f;
    for (int d = t; d < D_LIN;  d += 256) sn += fabsf(Wo[d * PCLS + c]);
    for (int d = t; d < D_CONV; d += 256) sn += fabsf(Wc[d * PCLS + c]);
    for (int d = t; d < D_REL;  d += 256) sr += fabsf(Wr[d * PCLS + c]);
    __shared__ float red[256];
    red[t] = sn; __syncthreads();
    for (int s = 128; s > 0; s >>= 1) { if (t < s) red[t] += red[t + s]; __syncthreads(); }
    if (t == 0) L1n[c] = red[0];
    __syncthreads();
    red[t] = sr; __syncthreads();
    for (int s = 128; s > 0; s >>= 1) { if (t < s) red[t] += red[t + s]; __syncthreads(); }
    if (t == 0) L1r[c] = red[0];
}

// ---------------- kernel: node = obj_lin@Wobj + conv@Wconv (f32 WMMA) --------
// grid: NNODE/16 blocks of 128 threads (4 waves). Wave w -> N-tile [16w,16w+16).
// A-panel (16 rows x 64 K) staged in LDS via async copies, shared by all waves.
__global__ __launch_bounds__(128)
void node_gemm_kernel(const float* __restrict__ obj_lin,
                      const float* __restrict__ obj_conv,
                      const float* __restrict__ Wp_obj,
                      const float* __restrict__ Wp_conv,
                      float* __restrict__ node) {
    __shared__ __align__(16) float sA[16 * NSTRIDE];

    const int t    = threadIdx.x;
    const int lane = t & 31;
    const int n0   = (t >> 5) << 4;                // 0,16,32,48
    const int row0 = blockIdx.x << 4;              // 16 nodes per block
    const int r    = lane & 15;                    // row (A) / col-in-tile (B,C)
    const int ksel = (lane >> 4) << 1;             // 0 for lanes 0-15, 2 for 16-31

    v8f acc = {};

    // two GEMM panels: {obj_lin, ld=1024, Wp_obj}, {obj_conv, ld=12544, Wp_conv}
    for (int panel = 0; panel < 2; ++panel) {
        const float* mat = panel ? obj_conv : obj_lin;
        const float* Wp  = panel ? Wp_conv  : Wp_obj;
        const int    ld  = panel ? D_CONV   : D_LIN;
        const int    Kt  = panel ? D_CONV   : D_LIN;

        for (int k0 = 0; k0 < Kt; k0 += NCHUNK) {
            // stage 16 x 64 A-tile: 256 float4 across 128 threads (2 each)
            #pragma unroll
            for (int q = 0; q < 2; ++q) {
                const int i   = q * 128 + t;       // 0..255
                const int row = i >> 4;            // 16 float4 per row
                const int c4  = (i & 15) << 2;
                const unsigned loff =
                    (unsigned)(uintptr_t)&sA[row * NSTRIDE + c4];
                async_g2l_b128(loff, mat + (size_t)(row0 + row) * ld + k0 + c4);
            }
            wait_asynccnt0();
            __syncthreads();

            // 16 WMMA steps out of LDS (banks (4r+k) mod 64 -> conflict-free)
            #pragma unroll 4
            for (int kl = 0; kl < NCHUNK; kl += 4) {
                v2f a; a.x = sA[r * NSTRIDE + kl + ksel];
                       a.y = sA[r * NSTRIDE + kl + ksel + 1];
                const int kk = k0 + kl + ksel;
                v2f b; b.x = Wp[kk * PPAD + n0 + r];
                       b.y = Wp[(kk + 1) * PPAD + n0 + r];
                acc = __builtin_amdgcn_wmma_f32_16x16x4_f32(false, a, false, b,
                                                            (short)0, acc, false, false);
            }
            __syncthreads();                       // protect sA before restage
        }
    }

    // ---- store C: lane holds rows v + 8*(lane>=16), col n0+r ----
    const int rowoff = (lane >> 4) << 3;
    #pragma unroll
    for (int v = 0; v < 8; ++v)
        node[(size_t)(row0 + rowoff + v) * PPAD + n0 + r] = acc[v];
}

// ---------------- kernel: edge GEMM + gather + per-class argmax --------------
// grid: NEDGE/64 blocks of 128 threads; each block = 4 M-tiles of one graph.
// Full 16x256 A-panel per M-tile staged in LDS via async copies.
__global__ __launch_bounds__(128)
void edge_max_kernel(const float* __restrict__ rel_lin,
                     const float* __restrict__ Wp_rel,
                     const float* __restrict__ node,
                     const int*   __restrict__ ridx,
                     unsigned long long* __restrict__ enc) {
    __shared__ __align__(16) float sA[16 * ESTRIDE];

    const int t    = threadIdx.x;
    const int lane = t & 31;
    const int n0   = (t >> 5) << 4;
    const int r    = lane & 15;
    const int ksel = (lane >> 4) << 1;
    const int eb0  = blockIdx.x << 6;              // 64 edges, same graph

    float    best  = -3.4e38f;
    unsigned bidx  = 0u;

    for (int m = 0; m < 4; ++m) {
        const int e0 = eb0 + (m << 4);

        // stage 16 x 256 A-panel: 1024 float4 across 128 threads (8 each)
        #pragma unroll
        for (int q = 0; q < 8; ++q) {
            const int i   = q * 128 + t;           // 0..1023
            const int row = i >> 6;                // 64 float4 per row
            const int c4  = (i & 63) << 2;
            const unsigned loff = (unsigned)(uintptr_t)&sA[row * ESTRIDE + c4];
            async_g2l_b128(loff, rel_lin + (size_t)(e0 + row) * D_REL + c4);
        }
        wait_asynccnt0();
        __syncthreads();

        v8f acc = {};
        #pragma unroll 4
        for (int kl = 0; kl < D_REL; kl += 4) {
            v2f a; a.x = sA[r * ESTRIDE + kl + ksel];
                   a.y = sA[r * ESTRIDE + kl + ksel + 1];
            const int kk = kl + ksel;
            v2f b; b.x = Wp_rel[kk * PPAD + n0 + r];
                   b.y = Wp_rel[(kk + 1) * PPAD + n0 + r];
            acc = __builtin_amdgcn_wmma_f32_16x16x4_f32(false, a, false, b,
                                                        (short)0, acc, false, false);
        }

        const int rowoff = (lane >> 4) << 3;
        #pragma unroll
        for (int v = 0; v < 8; ++v) {
            const int e  = e0 + rowoff + v;
            const int sn = ridx[e];
            const int dn = ridx[NEDGE + e];
            const float val = acc[v] + node[(size_t)sn * PPAD + n0 + r]
                                     + node[(size_t)dn * PPAD + n0 + r];
            if (val > best) { best = val; bidx = (unsigned)e; }
        }
        __syncthreads();                           // protect sA before restage
    }

    // combine the two lane halves holding the same class (rows 0-7 vs 8-15)
    const float    ob = __shfl_xor(best, 16, 32);
    const unsigned oi = __shfl_xor(bidx, 16, 32);
    if (ob > best || (ob == best && oi < bidx)) { best = ob; bidx = oi; }

    if (lane < 16) {
        const int b = eb0 >> 11;                   // / EPG
        const unsigned long long e =
            ((unsigned long long)f2ord(best) << 32) |
            (unsigned long long)(~bidx);           // ~idx => lower idx wins ties
        atomicMax(&enc[b * PPAD + n0 + r], e);
    }
}

// ---------------- kernel: sigmoid, top-K, analytic relevance, top-X ----------
__global__ void finalize_kernel(const unsigned long long* __restrict__ enc,
                                const float* __restrict__ L1n,
                                const float* __restrict__ L1r,
                                const int*   __restrict__ ridx,
                                float* __restrict__ out) {
    const int b = blockIdx.x;
    const int t = threadIdx.x;
    __shared__ float    sprob[PCLS];
    __shared__ unsigned sedge[PCLS];
    if (t < PCLS) {
        const unsigned long long e = enc[b * PPAD + t];
        const float logit = ord2f((unsigned)(e >> 32));
        sprob[t] = 1.0f / (1.0f + expf(-logit));
        sedge[t] = ~(unsigned)(e & 0xFFFFFFFFull);
    }
    __syncthreads();
    if (t != 0) return;

    // ---- top-10 classes (ties: lowest class index first, like lax.top_k) ----
    float pks[KTOP]; int pkc[KTOP];
    bool used[PCLS];
    for (int c = 0; c < PCLS; ++c) used[c] = false;
    for (int k = 0; k < KTOP; ++k) {
        float bv = -1.0f; int bc = 0;
        for (int c = 0; c < PCLS; ++c)
            if (!used[c] && sprob[c] > bv) { bv = sprob[c]; bc = c; }
        used[bc] = true; pks[k] = bv; pkc[k] = bc;
    }

    // ---- analytic relevance: relation_scores is one-hot per (b,k) ----
    float cs[KTOP]; int cf[KTOP];
    for (int k = 0; k < KTOP; ++k) {
        const int   c  = pkc[k];
        const float s  = pks[k];
        const float g  = s * (1.0f - s);
        const unsigned eg = sedge[c];
        const int sn = ridx[eg], dn = ridx[NEDGE + eg];
        const float coef = (sn == dn) ? 2.0f * g : g;  // self-loop doubles node grad
        const float rnv  = coef * L1n[c];
        const float re   = g * L1r[c];
        cs[k] = rnv * re * rnv * s;
        cf[k] = (int)(eg - (unsigned)(b * EPG)) * KTOP + k;
    }
    // insertion sort: score desc, flat-idx asc on ties
    for (int i = 1; i < KTOP; ++i) {
        float s = cs[i]; int f = cf[i]; int j = i - 1;
        while (j >= 0 && (cs[j] < s || (cs[j] == s && cf[j] > f))) {
            cs[j + 1] = cs[j]; cf[j + 1] = cf[j]; --j;
        }
        cs[j + 1] = s; cf[j + 1] = f;
    }

    const int obase = b * XTOP;
    auto emit = [&](int j, float score, int flat) {
        const int el = flat / KTOP, kk = flat % KTOP;
        const int eg = b * EPG + el;
        out[obase + j]                         = score;                   // scores
        out[BGRAPH * XTOP     + obase + j]     = (float)ridx[eg];         // src row
        out[BGRAPH * XTOP * 2 + obase + j]     = (float)ridx[NEDGE + eg]; // dst row
        out[BGRAPH * XTOP * 3 + obase + j]     = pks[kk];                 // pred score
        out[BGRAPH * XTOP * 4 + obase + j]     = (float)pkc[kk];          // pred class
    };

    int j = 0;
    for (; j < KTOP; ++j) emit(j, cs[j], cf[j]);
    // remaining 90: zero scores at ascending flat indices, skipping candidates
    int f = 0;
    for (; j < XTOP; ++j) {
        for (;;) {
            bool isc = false;
            for (int q = 0; q < KTOP; ++q) if (cf[q] == f) { isc = true; break; }
            if (!isc) break;
            ++f;
        }
        emit(j, 0.0f, f);
        ++f;
    }
    out[BGRAPH * XTOP * 5 + b] = (float)XTOP;   // n_relations
}

// ---------------- launcher ----------------
extern "C" void kernel_launch(void* const* d_in, const int* in_sizes, int n_in,
                              void* d_out, int out_size, void* d_ws, size_t ws_size,
                              hipStream_t stream) {
    const float* obj_lin  = (const float*)d_in[0];
    const float* obj_conv = (const float*)d_in[1];
    const float* rel_lin  = (const float*)d_in[2];
    const int*   ridx     = (const int*)  d_in[3];
    const float* W_obj    = (const float*)d_in[4];
    const float* W_conv   = (const float*)d_in[5];
    const float* W_rel    = (const float*)d_in[6];
    float* out = (float*)d_out;

    // workspace layout (u64 first for alignment)
    char* ws = (char*)d_ws;
    size_t off = 0;
    unsigned long long* enc = (unsigned long long*)(ws + off); off += (size_t)BGRAPH * PPAD * 8;
    float* Wp_obj  = (float*)(ws + off); off += (size_t)D_LIN  * PPAD * 4;
    float* Wp_conv = (float*)(ws + off); off += (size_t)D_CONV * PPAD * 4;
    float* Wp_rel  = (float*)(ws + off); off += (size_t)D_REL  * PPAD * 4;
    float* node    = (float*)(ws + off); off += (size_t)NNODE  * PPAD * 4;
    float* L1n     = (float*)(ws + off); off += PPAD * 4;
    float* L1r     = (float*)(ws + off); off += PPAD * 4;

    // 1) prep: init argmax slots, pad weights, L1 norms
    init_enc_kernel<<<(BGRAPH * PPAD + 255) / 256, 256, 0, stream>>>(enc, BGRAPH * PPAD);
    pad_weights_kernel<<<(D_LIN  * PPAD + 255) / 256, 256, 0, stream>>>(W_obj,  Wp_obj,  D_LIN  * PPAD);
    pad_weights_kernel<<<(D_CONV * PPAD + 255) / 256, 256, 0, stream>>>(W_conv, Wp_conv, D_CONV * PPAD);
    pad_weights_kernel<<<(D_REL  * PPAD + 255) / 256, 256, 0, stream>>>(W_rel,  Wp_rel,  D_REL  * PPAD);
    l1_kernel<<<PCLS, 256, 0, stream>>>(W_obj, W_conv, W_rel, L1n, L1r);

    // 2) node features GEMM (fp32 WMMA, async-LDS staged A panels)
    node_gemm_kernel<<<NNODE / 16, 128, 0, stream>>>(obj_lin, obj_conv, Wp_obj, Wp_conv, node);

    // 3) edge GEMM + gather + per-(graph,class) max/argmax
    edge_max_kernel<<<NEDGE / 64, 128, 0, stream>>>(rel_lin, Wp_rel, node, ridx, enc);

    // 4) sigmoid/top-K/analytic backward/top-X assembly
    finalize_kernel<<<BGRAPH, 64, 0, stream>>>(enc, L1n, L1r, ridx, out);
}